// LossHardNegativeMining_15719580303995
// MI455X (gfx1250) — compile-verified
//
#include <hip/hip_runtime.h>
#include <hip/hip_bf16.h>
#include <stdint.h>

// Problem geometry (fixed by reference setup_inputs)
#define NN     8
#define CC     16
#define HWSZ   65536                 // 256*256
#define LTOT   (NN*HWSZ)             // 524288 elements per channel group
#define TOTAL  (NN*CC*HWSZ)          // 8388608 elements
#define IGNN   10                    // ignore_largest_n
#define THRESH 0.1f
#define HBINS  8192                  // per-group histogram bins over (0,1)
#define CHUNK  16384                 // elements per block (within one (n,c) plane)
#define TILE   4096                  // TDM tile (floats) staged into LDS
#define NTILE  (CHUNK/TILE)

typedef unsigned int v4u __attribute__((ext_vector_type(4)));
typedef int          v8i __attribute__((ext_vector_type(8)));
typedef int          v4i __attribute__((ext_vector_type(4)));
typedef float        v4f __attribute__((ext_vector_type(4)));

// ---------------------------------------------------------------------------
// TDM: async 1-D tile load Global -> LDS via Tensor Data Mover (CDNA5).
// Descriptor per CDNA5 ISA:
//   group0: count=1 | lds_addr | global_addr[56:0] | type=2
//   group1: data_size=4B, tensor_dim0=nelem, tensor_dim1=1, tile_dim0=nelem,
//           tile_dim1=1 (single row), strides=0, no multicast/pad/iterate.
// Issued by one wave (EXEC ignored by TDM); completion via TENSORcnt.
// ---------------------------------------------------------------------------
__device__ __forceinline__ void tdm_load_f32_1d(unsigned lds_off, const void* gptr,
                                                unsigned nelem) {
  unsigned long long ga = (unsigned long long)gptr;
  v4u g0;
  g0.x = 1u;                                                  // count=1, user mode
  g0.y = lds_off;                                             // LDS byte address
  g0.z = (unsigned)(ga & 0xffffffffull);                      // global_addr[31:0]
  g0.w = (unsigned)((ga >> 32) & 0x1ffffffull) | (2u << 30);  // [56:32] | type=2
  v8i g1;
  g1[0] = (int)(2u << 16);                                    // data_size=2 -> 4 bytes
  g1[1] = (int)((nelem & 0xffffu) << 16);                     // tensor_dim0[15:0] @63:48
  g1[2] = (int)(((nelem >> 16) & 0xffffu) | (1u << 16));      // tensor_dim0 hi | tensor_dim1=1
  g1[3] = (int)((nelem & 0xffffu) << 16);                     // tile_dim0 @127:112
  g1[4] = 1;                                                  // tile_dim1=1, tile_dim2=0
  g1[5] = 0; g1[6] = 0; g1[7] = 0;                            // strides (single row)
  v4i z4 = {0, 0, 0, 0};
#if defined(__clang_major__) && (__clang_major__ >= 23)
  v8i z8 = {0, 0, 0, 0, 0, 0, 0, 0};
  __builtin_amdgcn_tensor_load_to_lds(g0, g1, z4, z4, z8, 0);
#else
  __builtin_amdgcn_tensor_load_to_lds(g0, g1, z4, z4, 0);
#endif
}

__device__ __forceinline__ unsigned hash32(unsigned x) {
  x ^= x >> 17; x *= 0xed5ad4bbu;
  x ^= x >> 11; x *= 0xac4c1b51u;
  x ^= x >> 15; x *= 0x31848babu;
  x ^= x >> 14;
  return x;
}

// ---------------------------------------------------------------------------
// Kernel 0: zero the histogram + counter region of the workspace each launch.
// ---------------------------------------------------------------------------
__global__ void __launch_bounds__(256) k_zero(unsigned* __restrict__ p, int n) {
  int i = blockIdx.x * 256 + threadIdx.x;
  if (i < n) p[i] = 0u;
}

// ---------------------------------------------------------------------------
// Kernel 1: streaming pass. TDM double-buffers loss tiles into LDS
// (tensor_load_to_lds + s_wait_tensorcnt); mask read with coalesced b128
// loads; loss consumed from LDS via conflict-free ds_load_b128 (thread
// stride 16B across 64 banks). Per-group LDS histogram (ds_add atomics) and
// positive / non-positive counts, flushed with global atomics.
// ---------------------------------------------------------------------------
__global__ void __launch_bounds__(256) k_stats(const float* __restrict__ loss,
                                               const int* __restrict__ tmask,
                                               unsigned* __restrict__ hist,
                                               unsigned* __restrict__ posc,
                                               unsigned* __restrict__ nonc) {
  extern __shared__ unsigned char smem[];
  float*    tbuf  = (float*)smem;                              // 2*TILE floats (32 KB)
  unsigned* shist = (unsigned*)(smem + 2 * TILE * 4);          // HBINS u32    (32 KB)
  unsigned* sred  = (unsigned*)(smem + 2 * TILE * 4 + HBINS * 4);

  const int chunkBase = blockIdx.x * CHUNK;
  const int c = (chunkBase / HWSZ) % CC;

  for (int b = threadIdx.x; b < HBINS; b += 256) shist[b] = 0u;
  if (threadIdx.x == 0) { sred[0] = 0u; sred[1] = 0u; }

  const bool w0 = (threadIdx.x < 32);   // wave 0 owns TDM issue/wait (wave32)
  if (w0) tdm_load_f32_1d(0u, loss + chunkBase, TILE);

  unsigned pc = 0u, nc = 0u;
  for (int k = 0; k < NTILE; ++k) {
    if (w0) {
      if (k + 1 < NTILE) {   // prefetch next tile into the other buffer
        tdm_load_f32_1d((unsigned)(((k + 1) & 1) * TILE * 4),
                        loss + chunkBase + (k + 1) * TILE, TILE);
        __builtin_amdgcn_s_wait_tensorcnt(1);   // tile k complete (in-order TDM)
      } else {
        __builtin_amdgcn_s_wait_tensorcnt(0);
      }
    }
    __syncthreads();                       // tile k visible to all waves
    const float* tl   = tbuf + (k & 1) * TILE;
    const int    gbas = chunkBase + k * TILE;
#pragma unroll
    for (int sub = 0; sub < TILE / 1024; ++sub) {   // 4 x (4-wide) per thread
      int e  = threadIdx.x * 4 + sub * 1024;
      v4f lv = *(const v4f*)(tl + e);               // ds_load_b128
      v4i tm = *(const v4i*)(tmask + gbas + e);     // global_load_b128
#pragma unroll
      for (int j = 0; j < 4; ++j) {
        float lvv = lv[j];
        int   tmm = tm[j];
        // l = loss*(loss>=T)*(tm==NEG) - (tm==IGNORE)
        float l = (tmm == 0 && lvv >= THRESH) ? lvv : 0.0f;
        if (tmm == -1) l = -1.0f;
        pc += (tmm == 1) ? 1u : 0u;
        if (l <= 0.0f) {
          nc += 1u;
        } else {
          int b = (int)(l * (float)HBINS);
          b = (b > HBINS - 1) ? (HBINS - 1) : b;
          atomicAdd(&shist[b], 1u);
        }
      }
    }
    __syncthreads();                       // done reading buffer before reuse
  }

  atomicAdd(&sred[0], pc);
  atomicAdd(&sred[1], nc);
  __syncthreads();
  if (threadIdx.x == 0) {
    atomicAdd(&posc[c], sred[0]);
    atomicAdd(&nonc[c], sred[1]);
  }
  unsigned* gh = hist + c * HBINS;
  for (int b = threadIdx.x; b < HBINS; b += 256) {
    unsigned v = shist[b];
    if (v) atomicAdd(&gh[b], v);
  }
}

// ---------------------------------------------------------------------------
// Kernel 2: one block per group. Prefix-scan the histogram, convert the three
// rank cutoffs (10, 10+hard_remain, total_count) of the descending order into
// value thresholds tA > tB > tC with within-bin interpolation, and compute the
// Bernoulli probability for the random-negative window.
// ---------------------------------------------------------------------------
__global__ void __launch_bounds__(256) k_params(const unsigned* __restrict__ hist,
                                                const unsigned* __restrict__ posc,
                                                const unsigned* __restrict__ nonc,
                                                unsigned* __restrict__ params) {
  __shared__ unsigned part[256];
  __shared__ unsigned basex[256];
  __shared__ float    res[3];
  __shared__ unsigned totS;

  const int c = blockIdx.x;
  const unsigned* h = hist + c * HBINS;
  const int t = threadIdx.x;

  unsigned lb[HBINS / 256];
  unsigned s = 0u;
#pragma unroll
  for (int j = 0; j < HBINS / 256; ++j) { lb[j] = h[t * (HBINS / 256) + j]; s += lb[j]; }
  part[t] = s;
  if (t < 3) res[t] = 0.0f;
  __syncthreads();
  if (t == 0) {
    unsigned run0 = 0u;
    for (int i = 0; i < 256; ++i) { basex[i] = run0; run0 += part[i]; }
    totS = run0;
  }
  __syncthreads();

  const int Tot  = (int)totS;                 // #elements with l > 0
  const int cnt  = (int)posc[c];
  const int ignc = (int)nonc[c];
  int total_count = LTOT - IGNN - ignc; if (total_count < 0) total_count = 0;
  // total_remain = (cnt/0.5).int - cnt == cnt; then max(.,1), min(.,tc), max(.,0)
  int tr = cnt; if (tr < 1) tr = 1;
  if (tr > total_count) tr = total_count;
  if (tr < 0) tr = 0;
  int hard   = tr >> 1;                       // floor(0.5 * total_remain)
  int normal = tr - hard;
  int valid  = (total_count >= 1 && tr >= 1) ? 1 : 0;
  if (!valid) { hard = 0; normal = 0; }

  int rr[3];  rr[0] = IGNN; rr[1] = IGNN + hard; rr[2] = total_count;
  int tgt[3];
  for (int q = 0; q < 3; ++q) tgt[q] = Tot - rr[q];   // prefix-from-bottom target

  unsigned run = basex[t];
#pragma unroll
  for (int j = 0; j < HBINS / 256; ++j) {
    unsigned cb   = lb[j];
    unsigned pnew = run + cb;
    if (cb) {
      int b = t * (HBINS / 256) + j;
      for (int q = 0; q < 3; ++q) {
        int tg = tgt[q];
        if (tg > (int)run && tg <= (int)pnew) {
          float f = (float)((int)pnew - tg) / (float)cb;      // within-bin fraction
          res[q] = ((float)(b + 1) - f) / (float)HBINS;       // interpolated value
        }
      }
    }
    run = pnew;
  }
  __syncthreads();
  if (t == 0) {
    float tA = res[0], tB = res[1], tC = res[2];
    int numCand = total_count - (IGNN + hard); if (numCand < 0) numCand = 0;
    float p = 0.0f;
    if (valid && numCand > 0 && normal > 0) {
      p = (float)normal / (float)numCand;
      if (p > 1.0f) p = 1.0f;
    }
    unsigned randT = (unsigned)(p * 4294967040.0f);
    unsigned* P = params + c * 8;
    P[0] = __float_as_uint(tA);
    P[1] = __float_as_uint(tB);
    P[2] = __float_as_uint(tC);
    P[3] = randT;
    P[4] = (unsigned)valid;
    P[5] = 0u; P[6] = 0u; P[7] = 0u;
  }
}

// ---------------------------------------------------------------------------
// Kernel 3: final elementwise mask. Inputs are L2-resident after pass 1.
// out = (tm==POS) | (tB < l <= tA)                       // hard negatives
//                 | (tC < l <= tB  &&  hash(i) < randT)  // sampled negatives
// Output streamed with non-temporal b128 stores (never re-read).
// ---------------------------------------------------------------------------
__global__ void __launch_bounds__(256) k_final(const float* __restrict__ loss,
                                               const int* __restrict__ tmask,
                                               const unsigned* __restrict__ params,
                                               float* __restrict__ out) {
  const int chunkBase = blockIdx.x * CHUNK;
  const int c = (chunkBase / HWSZ) % CC;
  const unsigned* P = params + c * 8;
  const float    tA    = __uint_as_float(P[0]);
  const float    tB    = __uint_as_float(P[1]);
  const float    tC    = __uint_as_float(P[2]);
  const unsigned randT = P[3];
  const int      valid = (int)P[4];

#pragma unroll 4
  for (int it = 0; it < CHUNK / 1024; ++it) {       // 16 iterations of 4-wide work
    int i0 = chunkBase + it * 1024 + threadIdx.x * 4;
    v4f lv = *(const v4f*)(loss + i0);              // global_load_b128
    v4i tm = *(const v4i*)(tmask + i0);             // global_load_b128
    v4f ov;
#pragma unroll
    for (int j = 0; j < 4; ++j) {
      float lvv = lv[j];
      int   tmm = tm[j];
      float l  = (tmm == 0 && lvv >= THRESH) ? lvv : 0.0f;
      bool sel = (tmm == 1);                                  // keep_pos
      if (valid && l > 0.0f) {
        if (l <= tA && l > tB)       sel = true;              // hard window
        else if (l <= tB && l > tC)  sel = sel | (hash32((unsigned)(i0 + j)) < randT);
      }
      ov[j] = sel ? 1.0f : 0.0f;
    }
    __builtin_nontemporal_store(ov, (v4f*)(out + i0));        // b128 NT store
  }
}

// ---------------------------------------------------------------------------
// Launch: zero -> stats(+TDM) -> params -> final, all on `stream`.
// Workspace layout (u32): hist[16*8192] | posc[16] | nonc[16] | params[16*8]
// (~513 KB). Histogram/counters re-zeroed every call for determinism.
// ---------------------------------------------------------------------------
extern "C" void kernel_launch(void* const* d_in, const int* in_sizes, int n_in,
                              void* d_out, int out_size, void* d_ws, size_t ws_size,
                              hipStream_t stream) {
  const float* loss  = (const float*)d_in[0];
  const int*   tmask = (const int*)d_in[1];
  float*       out   = (float*)d_out;

  unsigned* ws     = (unsigned*)d_ws;
  unsigned* hist   = ws;
  unsigned* posc   = ws + CC * HBINS;
  unsigned* nonc   = posc + CC;
  unsigned* params = nonc + CC;

  const int nz = CC * HBINS + 2 * CC;
  k_zero<<<(nz + 255) / 256, 256, 0, stream>>>(ws, nz);

  const size_t shmem = 2 * TILE * sizeof(float) + HBINS * sizeof(unsigned) + 16;
  k_stats<<<TOTAL / CHUNK, 256, shmem, stream>>>(loss, tmask, hist, posc, nonc);
  k_params<<<CC, 256, 0, stream>>>(hist, posc, nonc, params);
  k_final<<<TOTAL / CHUNK, 256, 0, stream>>>(loss, tmask, params, out);
}